// SetAbstraction_89438398972566
// MI455X (gfx1250) — compile-verified
//
#include <hip/hip_runtime.h>
#include <hip/hip_bf16.h>

#define B_   2
#define N_   16384
#define S_   4096
#define K_   32
#define FIN  16
#define R2_  0.01f
#define NEGV (-1e30f)

typedef __attribute__((ext_vector_type(2))) float v2f;
typedef __attribute__((ext_vector_type(8))) float v8f;

// ------------------------------------------------------------------
// Kernel 1: farthest point sampling. One block per cloud, 1024 thr.
// Each thread owns 16 points in registers; argmax via wave shuffles
// + cross-wave LDS reduction. Serial over S-1 steps by necessity.
// ------------------------------------------------------------------
__global__ __launch_bounds__(1024, 1)
void fps_kernel(const float* __restrict__ pos, int* __restrict__ fpsIdx)
{
  const int b    = blockIdx.x;
  const int tid  = threadIdx.x;
  const int lane = tid & 31;
  const int wid  = tid >> 5;
  __shared__ float rv[32];
  __shared__ int   ri[32];
  __shared__ int   s_last;

  float px[16], py[16], pz[16], dmin[16];
#pragma unroll
  for (int i = 0; i < 16; ++i) {
    const int pid = tid + i * 1024;
    const float* p = pos + ((size_t)b * N_ + pid) * 3;
    px[i] = p[0]; py[i] = p[1]; pz[i] = p[2];
    dmin[i] = 1e10f;
  }
  if (tid == 0) { s_last = 0; fpsIdx[b * S_] = 0; }
  __syncthreads();

  for (int j = 1; j < S_; ++j) {
    const int last = s_last;
    const float* c = pos + ((size_t)b * N_ + last) * 3;
    const float cx = c[0], cy = c[1], cz = c[2];
    float bestv = -1.0f; int besti = 0;
#pragma unroll
    for (int i = 0; i < 16; ++i) {
      const float dx = px[i] - cx, dy = py[i] - cy, dz = pz[i] - cz;
      const float d = dx * dx + dy * dy + dz * dz;
      const float dm = dmin[i] < d ? dmin[i] : d;
      dmin[i] = dm;
      if (dm > bestv) { bestv = dm; besti = tid + i * 1024; }
    }
#pragma unroll
    for (int off = 16; off >= 1; off >>= 1) {
      const float ov = __shfl_xor(bestv, off, 32);
      const int   oi = __shfl_xor(besti, off, 32);
      if (ov > bestv || (ov == bestv && oi < besti)) { bestv = ov; besti = oi; }
    }
    if (lane == 0) { rv[wid] = bestv; ri[wid] = besti; }
    __syncthreads();
    if (wid == 0) {
      bestv = rv[lane]; besti = ri[lane];
#pragma unroll
      for (int off = 16; off >= 1; off >>= 1) {
        const float ov = __shfl_xor(bestv, off, 32);
        const int   oi = __shfl_xor(besti, off, 32);
        if (ov > bestv || (ov == bestv && oi < besti)) { bestv = ov; besti = oi; }
      }
      if (lane == 0) { s_last = besti; fpsIdx[b * S_ + j] = besti; }
    }
    __syncthreads();
  }
}

// ------------------------------------------------------------------
// Kernel 2: gather centers, emit batch_out.
// ------------------------------------------------------------------
__global__ void centers_kernel(const float* __restrict__ pos, const int* __restrict__ fpsIdx,
                               float* __restrict__ cen, float* __restrict__ bat)
{
  const int i = blockIdx.x * blockDim.x + threadIdx.x;
  if (i < B_ * S_) {
    const int b = i / S_;
    const int idx = fpsIdx[i];
    const float* p = pos + ((size_t)b * N_ + idx) * 3;
    cen[i * 3 + 0] = p[0];
    cen[i * 3 + 1] = p[1];
    cen[i * 3 + 2] = p[2];
    bat[i] = (float)b;
  }
}

// ------------------------------------------------------------------
// Kernel 3: ball query. One wave per centroid; ballot/popc stream
// compaction to take the first K in-radius points in index order.
// Self-loop removed when b*N+n == b*S+sc (PyG numeric quirk).
// ------------------------------------------------------------------
__global__ __launch_bounds__(128, 1)
void ballq_kernel(const float* __restrict__ pos, const float* __restrict__ centers,
                  int* __restrict__ nbr, int* __restrict__ cnts)
{
  const int lane = threadIdx.x & 31;
  const int wid  = threadIdx.x >> 5;
  const int cid  = blockIdx.x * 4 + wid;          // 0..8191
  const int b = cid / S_, sc = cid % S_;
  __shared__ int snbr[4][K_];
  snbr[wid][lane] = 0;
  const float cx = centers[cid * 3 + 0];
  const float cy = centers[cid * 3 + 1];
  const float cz = centers[cid * 3 + 2];
  const int self_n = b * S_ + sc - b * N_;        // negative for b>0 -> never matches
  int cnt = 0;
  for (int base = 0; base < N_ && cnt < K_; base += 32) {
    const int n = base + lane;
    const float* p = pos + ((size_t)b * N_ + n) * 3;
    const float dx = p[0] - cx, dy = p[1] - cy, dz = p[2] - cz;
    const float d2 = dx * dx + dy * dy + dz * dz;
    const bool act = (d2 <= R2_) && (n != self_n);
    const unsigned mask = (unsigned)__ballot(act);
    const int pre = __popc(mask & ((1u << lane) - 1u));
    const int slot = cnt + pre;
    if (act && slot < K_) snbr[wid][slot] = n;
    cnt += __popc(mask);
  }
  if (cnt > K_) cnt = K_;
  nbr[cid * K_ + lane] = snbr[wid][lane];
  if (lane == 0) cnts[cid] = cnt;
}

// ------------------------------------------------------------------
// Kernel 4: fused 19->64->64->128 MLP + masked max, fp32 WMMA.
// One wave per centroid; 33 rows padded to 48 (3 M-tiles of 16).
// Weights in workgroup LDS; feats/activations in wave-private LDS.
// ------------------------------------------------------------------
__global__ __launch_bounds__(128, 1)
void mlp_kernel(const float* __restrict__ x, const float* __restrict__ pos,
                const float* __restrict__ W1, const float* __restrict__ b1,
                const float* __restrict__ W2, const float* __restrict__ b2,
                const float* __restrict__ W3, const float* __restrict__ b3,
                const float* __restrict__ centers, const int* __restrict__ nbr,
                const int* __restrict__ cnts, float* __restrict__ xout)
{
  extern __shared__ float sm[];
  float* sW1 = sm;                  // [20][64]  (row 19 zero pad)
  float* sW2 = sW1 + 20 * 64;       // [64][64]
  float* sW3 = sW2 + 64 * 64;       // [64][128]
  float* sb1 = sW3 + 64 * 128;      // [64]
  float* sb2 = sb1 + 64;            // [64]
  float* sb3 = sb2 + 64;            // [128]
  float* sWv = sb3 + 128;
  const int lane = threadIdx.x & 31;
  const int wid  = threadIdx.x >> 5;
  float* F = sWv + wid * (48 * 20 + 48 * 65);   // [48][20]
  float* H = F + 48 * 20;                        // [48][65] padded stride

  for (int i = threadIdx.x; i < 20 * 64; i += 128) sW1[i] = (i < 19 * 64) ? W1[i] : 0.f;
  for (int i = threadIdx.x; i < 64 * 64; i += 128) sW2[i] = W2[i];
  for (int i = threadIdx.x; i < 64 * 128; i += 128) sW3[i] = W3[i];
  if (threadIdx.x < 64)  { sb1[threadIdx.x] = b1[threadIdx.x]; sb2[threadIdx.x] = b2[threadIdx.x]; }
  if (threadIdx.x < 128) sb3[threadIdx.x] = b3[threadIdx.x];
  __syncthreads();

  const int cid = blockIdx.x * 4 + wid;
  const int b = cid / S_, sc = cid % S_;
  const int cnt = cnts[cid];
  const float cx = centers[cid * 3 + 0];
  const float cy = centers[cid * 3 + 1];
  const float cz = centers[cid * 3 + 2];

  // ---- build feature rows: 0..31 neighbors, 32 self-loop, 33..47 zero ----
  {
    const int k = lane;
    const int g = nbr[cid * K_ + k];
    if (k < cnt) {
      const float* xr = x + ((size_t)b * N_ + g) * FIN;
#pragma unroll
      for (int c = 0; c < 16; ++c) F[k * 20 + c] = xr[c];
      const float* pr = pos + ((size_t)b * N_ + g) * 3;
      F[k * 20 + 16] = pr[0] - cx;
      F[k * 20 + 17] = pr[1] - cy;
      F[k * 20 + 18] = pr[2] - cz;
      F[k * 20 + 19] = 0.f;
    } else {
#pragma unroll
      for (int c = 0; c < 20; ++c) F[k * 20 + c] = 0.f;
    }
    if (lane == 0) {
      const int d = b * S_ + sc;   // PyG add_self_loops flat index quirk
      const float* xr = x + (size_t)d * FIN;
#pragma unroll
      for (int c = 0; c < 16; ++c) F[32 * 20 + c] = xr[c];
      const float* pr = pos + (size_t)d * 3;
      F[32 * 20 + 16] = pr[0] - cx;
      F[32 * 20 + 17] = pr[1] - cy;
      F[32 * 20 + 18] = pr[2] - cz;
      F[32 * 20 + 19] = 0.f;
    }
    if (lane >= 1 && lane <= 15) {
      const int r = 32 + lane;
#pragma unroll
      for (int c = 0; c < 20; ++c) F[r * 20 + c] = 0.f;
    }
  }

  const int ln   = lane & 15;
  const int hi   = (lane >> 4) & 1;
  const int koff = hi ? 2 : 0;       // A/B K-offset per ISA 16x16x4 f32 layout
  const int hr0  = hi * 8;           // C/D row offset per lane half
  const v8f vzero = {0.f, 0.f, 0.f, 0.f, 0.f, 0.f, 0.f, 0.f};

  // ---- Layer 1: F[48][20] @ W1[20][64] -> H (ReLU) ----
#pragma unroll
  for (int m = 0; m < 3; ++m) {
    const int row = m * 16 + ln;
#pragma unroll
    for (int t = 0; t < 4; ++t) {
      v8f acc = vzero;
#pragma unroll
      for (int kk = 0; kk < 5; ++kk) {
        const int k0 = kk * 4 + koff;
        v2f a, w;
        a.x = F[row * 20 + k0];
        a.y = F[row * 20 + k0 + 1];
        w.x = sW1[k0 * 64 + t * 16 + ln];
        w.y = sW1[(k0 + 1) * 64 + t * 16 + ln];
        acc = __builtin_amdgcn_wmma_f32_16x16x4_f32(false, a, false, w, (short)0, acc, false, false);
      }
      const float bias = sb1[t * 16 + ln];
#pragma unroll
      for (int j = 0; j < 8; ++j) {
        float v = acc[j] + bias;
        H[(m * 16 + hr0 + j) * 65 + t * 16 + ln] = v > 0.f ? v : 0.f;
      }
    }
  }

  // ---- Layer 2: H @ W2[64][64] -> H (ReLU), in-place per M-tile ----
#pragma unroll
  for (int m = 0; m < 3; ++m) {
    const int row = m * 16 + ln;
    v8f acc[4];
#pragma unroll
    for (int t = 0; t < 4; ++t) {
      acc[t] = vzero;
#pragma unroll
      for (int kk = 0; kk < 16; ++kk) {
        const int k0 = kk * 4 + koff;
        v2f a, w;
        a.x = H[row * 65 + k0];
        a.y = H[row * 65 + k0 + 1];
        w.x = sW2[k0 * 64 + t * 16 + ln];
        w.y = sW2[(k0 + 1) * 64 + t * 16 + ln];
        acc[t] = __builtin_amdgcn_wmma_f32_16x16x4_f32(false, a, false, w, (short)0, acc[t], false, false);
      }
    }
#pragma unroll
    for (int t = 0; t < 4; ++t) {
      const float bias = sb2[t * 16 + ln];
#pragma unroll
      for (int j = 0; j < 8; ++j) {
        float v = acc[t][j] + bias;
        H[(m * 16 + hr0 + j) * 65 + t * 16 + ln] = v > 0.f ? v : 0.f;
      }
    }
  }

  // ---- Layer 3: H @ W3[64][128] + b3, masked row-max -> out[128] ----
  const unsigned long long vmask =
      ((cnt >= 32) ? 0xFFFFFFFFull : ((1ull << cnt) - 1ull)) | (1ull << 32);
  float tmax[8];
#pragma unroll
  for (int t = 0; t < 8; ++t) tmax[t] = NEGV;
#pragma unroll
  for (int m = 0; m < 3; ++m) {
    const int row = m * 16 + ln;
#pragma unroll
    for (int t = 0; t < 8; ++t) {
      v8f acc = vzero;
#pragma unroll
      for (int kk = 0; kk < 16; ++kk) {
        const int k0 = kk * 4 + koff;
        v2f a, w;
        a.x = H[row * 65 + k0];
        a.y = H[row * 65 + k0 + 1];
        w.x = sW3[k0 * 128 + t * 16 + ln];
        w.y = sW3[(k0 + 1) * 128 + t * 16 + ln];
        acc = __builtin_amdgcn_wmma_f32_16x16x4_f32(false, a, false, w, (short)0, acc, false, false);
      }
      const float bias = sb3[t * 16 + ln];
      float pm = NEGV;
#pragma unroll
      for (int j = 0; j < 8; ++j) {
        const int r = m * 16 + hr0 + j;
        float v = acc[j] + bias;
        v = ((vmask >> r) & 1ull) ? v : NEGV;
        pm = fmaxf(pm, v);
      }
      tmax[t] = fmaxf(tmax[t], pm);
    }
  }
#pragma unroll
  for (int t = 0; t < 8; ++t) {
    const float o = __shfl_xor(tmax[t], 16, 32);
    const float v = fmaxf(tmax[t], o);
    if (lane < 16) xout[(size_t)cid * 128 + t * 16 + ln] = v;
  }
}

// ------------------------------------------------------------------
extern "C" void kernel_launch(void* const* d_in, const int* in_sizes, int n_in,
                              void* d_out, int out_size, void* d_ws, size_t ws_size,
                              hipStream_t stream)
{
  const float* x   = (const float*)d_in[0];
  const float* pos = (const float*)d_in[1];
  // d_in[2] = batch (recomputed, unused)
  const float* W1 = (const float*)d_in[3];
  const float* b1 = (const float*)d_in[4];
  const float* W2 = (const float*)d_in[5];
  const float* b2 = (const float*)d_in[6];
  const float* W3 = (const float*)d_in[7];
  const float* b3 = (const float*)d_in[8];

  float* out     = (float*)d_out;
  float* x_out   = out;                             // [B*S,128]
  float* cen_out = out + (size_t)B_ * S_ * 128;     // [B*S,3]
  float* bat_out = cen_out + (size_t)B_ * S_ * 3;   // [B*S]

  char* ws = (char*)d_ws;
  int* fpsIdx = (int*)ws;                 // B*S ints
  int* cnts   = (int*)(ws + 32768);       // B*S ints
  int* nbrs   = (int*)(ws + 65536);       // B*S*K ints

  fps_kernel<<<B_, 1024, 0, stream>>>(pos, fpsIdx);
  centers_kernel<<<(B_ * S_ + 255) / 256, 256, 0, stream>>>(pos, fpsIdx, cen_out, bat_out);
  ballq_kernel<<<B_ * S_ / 4, 128, 0, stream>>>(pos, cen_out, nbrs, cnts);
  const size_t smem = (size_t)(20 * 64 + 64 * 64 + 64 * 128 + 64 + 64 + 128 +
                               4 * (48 * 20 + 48 * 65)) * sizeof(float);
  mlp_kernel<<<B_ * S_ / 4, 128, smem, stream>>>(x, pos, W1, b1, W2, b2, W3, b3,
                                                 cen_out, nbrs, cnts, x_out);
}